// RDF_computer_4647154614876
// MI455X (gfx1250) — compile-verified
//
#include <hip/hip_runtime.h>
#include <math.h>

// ---------------------------------------------------------------------------
// RDF on MI455X (gfx1250, wave32).
//  * traj: [16, 512, 3] f32   cell: [3,3] f32   r_list: [118] f32
//  * out:  r_list (118) ++ gr (118)
// Strategy:
//  - enumerate i<j pairs directly (triangular index decode, no 2x waste)
//  - stage 128 pair distances per block tile in LDS
//  - bin-major ownership: each wave owns 2 groups of 16 bins
//  - Gaussian eval with v_exp_f32; pair-sum reduction done on the matrix
//    pipe via V_WMMA_F32_16X16X4_F32 with A == ones (column-sum trick),
//    C register accumulates across the whole run; one atomicAdd per
//    (block, bin) at the very end.
// ---------------------------------------------------------------------------

#define N_ATOMS   512
#define N_FRAMES  16
#define DR        0.1f
#define NBINS_PAD 128     // 8 groups of 16 bins (118 real + 10 padded)
#define TILE      128     // pair-distances staged per block iteration
#define BLOCK     128     // 4 wave32 waves
#define HIST_BLOCKS 1024

typedef __attribute__((ext_vector_type(2))) float v2f;
typedef __attribute__((ext_vector_type(8))) float v8f;

#ifndef __has_builtin
#define __has_builtin(x) 0
#endif
#if __has_builtin(__builtin_amdgcn_wmma_f32_16x16x4_f32)
#define RDF_USE_WMMA 1
#else
#define RDF_USE_WMMA 0
#endif

// number of pairs (a,b), a<b, with a < i  (row-start offset in linear pair id)
__device__ __forceinline__ int pairRowStart(int i) {
    return (i * (2 * N_ATOMS - 1 - i)) >> 1;
}

__global__ void rdf_zero_kernel(float* __restrict__ ws) {
    ws[threadIdx.x] = 0.0f;
}

__global__ __launch_bounds__(BLOCK) void rdf_hist_kernel(
    const float* __restrict__ traj,
    const float* __restrict__ cell,
    const float* __restrict__ r_list,
    int nbins,
    float* __restrict__ hist)
{
    __shared__ float s_d[TILE];
    __shared__ float s_r[NBINS_PAD];

    const int tid  = threadIdx.x;
    const int lane = tid & 31;
    const int wave = tid >> 5;

    // stage bin centers; padded bins get a far sentinel (gaussian == 0)
    s_r[tid] = (tid < nbins) ? r_list[tid] : 1.0e6f;

    const float Lx = cell[0], Ly = cell[4], Lz = cell[8];
    const float half = cell[0] * 0.5f;
    const float cutoff_sq = half * half;

    const int  P        = (N_ATOMS * (N_ATOMS - 1)) / 2;   // 130816
    const long total    = (long)N_FRAMES * (long)P;
    const long numTiles = (total + TILE - 1) / TILE;

    // this wave owns bin groups 2w and 2w+1 (16 bins each)
    const int g0 = wave * 2;
    const int g1 = wave * 2 + 1;
    const int n  = lane & 15;              // bin-within-group (B operand N = lane%16)
    const int kk = (lane >> 4) << 1;       // pair slots {0,1} or {2,3} within a 4-chunk

    __syncthreads();
    const float r0 = s_r[g0 * 16 + n];
    const float r1 = s_r[g1 * 16 + n];

#if RDF_USE_WMMA
    v8f c0 = {}; v8f c1 = {};
    const v2f onesA = {1.0f, 1.0f};
#else
    float a0 = 0.0f, a1 = 0.0f;
#endif

    for (long t = blockIdx.x; t < numTiles; t += gridDim.x) {
        // ---- phase 1: one pair distance per thread -> registers ----
        const long gk = t * (long)TILE + (long)tid;
        float d = 1.0e4f;                  // sentinel: gaussian underflows to 0
        if (gk < total) {
            const int f = (int)(gk / (long)P);
            const int k = (int)(gk - (long)f * (long)P);
            // decode triangular linear index -> (i, j), i < j
            const float tn   = 2.0f * (float)N_ATOMS - 1.0f;
            const float disc = tn * tn - 8.0f * (float)k;
            int i = (int)((tn - sqrtf(disc)) * 0.5f);
            i = i < 0 ? 0 : (i > N_ATOMS - 2 ? N_ATOMS - 2 : i);
            while (pairRowStart(i + 1) <= k) ++i;   // float-estimate fixup
            while (pairRowStart(i) > k)      --i;
            const int j = i + 1 + (k - pairRowStart(i));

            const float* qi = traj + ((long)f * N_ATOMS + i) * 3;
            const float* qj = traj + ((long)f * N_ATOMS + j) * 3;
            float dx = qi[0] - qj[0];
            float dy = qi[1] - qj[1];
            float dz = qi[2] - qj[2];
            dx -= floorf(dx / Lx + 0.5f) * Lx;       // minimum image
            dy -= floorf(dy / Ly + 0.5f) * Ly;
            dz -= floorf(dz / Lz + 0.5f) * Lz;
            const float sq = dx * dx + dy * dy + dz * dz;
            if (sq < cutoff_sq && sq != 0.0f) d = sqrtf(sq);
        }
        __syncthreads();                   // previous tile fully consumed
        s_d[tid] = d;
        __syncthreads();

        // ---- phase 2: bin-major gaussian eval; matrix-pipe column sums ----
        #pragma unroll 4
        for (int cix = 0; cix < TILE / 4; ++cix) {
            const float d0 = s_d[cix * 4 + kk];
            const float d1 = s_d[cix * 4 + kk + 1];
            const float t00 = (d0 - r0) * 10.0f;   // (d - r) / DR
            const float t10 = (d1 - r0) * 10.0f;
            const float t01 = (d0 - r1) * 10.0f;
            const float t11 = (d1 - r1) * 10.0f;
            const float e00 = __expf(-0.5f * t00 * t00);
            const float e10 = __expf(-0.5f * t10 * t10);
            const float e01 = __expf(-0.5f * t01 * t01);
            const float e11 = __expf(-0.5f * t11 * t11);
#if RDF_USE_WMMA
            v2f b0 = {e00, e10};
            v2f b1 = {e01, e11};
            // D = ones(16x4) * B(4x16) + C  ->  every row of D holds the
            // per-bin (column) partial sums; C accumulates across the run.
            c0 = __builtin_amdgcn_wmma_f32_16x16x4_f32(
                     false, onesA, false, b0, (short)0, c0, false, false);
            c1 = __builtin_amdgcn_wmma_f32_16x16x4_f32(
                     false, onesA, false, b1, (short)0, c1, false, false);
#else
            a0 += e00 + e10;
            a1 += e01 + e11;
#endif
        }
    }

    // ---- drain register accumulators to the global histogram ----
#if RDF_USE_WMMA
    if (lane < 16) {                       // row M=0 lives in component 0, lanes 0..15
        const int b0i = g0 * 16 + lane;
        const int b1i = g1 * 16 + lane;
        if (b0i < nbins) atomicAdd(&hist[b0i], c0[0]);
        if (b1i < nbins) atomicAdd(&hist[b1i], c1[0]);
    }
#else
    {   // both half-waves hold partials for bin (g*16 + lane%16)
        const int b0i = g0 * 16 + n;
        const int b1i = g1 * 16 + n;
        if (b0i < nbins) atomicAdd(&hist[b0i], a0);
        if (b1i < nbins) atomicAdd(&hist[b1i], a1);
    }
#endif
}

__global__ void rdf_finalize_kernel(
    const float* __restrict__ r_list,
    const float* __restrict__ cell,
    const float* __restrict__ ws,
    float* __restrict__ out,
    int nbins)
{
    const int b = blockIdx.x * blockDim.x + threadIdx.x;
    if (b >= nbins) return;
    const float r = r_list[b];
    // 1/(DR*sqrt(2*pi)) * DR / N_FRAMES applied here (factored out of the sum)
    const float GNORM = 3.9894228040143268f;   // 1 / (0.1 * sqrt(2*pi))
    const float h = ws[b] * GNORM * DR / (float)N_FRAMES;
    const float rp = r + 0.5f * DR, rm = r - 0.5f * DR;
    const float v = (4.0f * 3.14159265358979323846f / 3.0f)
                    * (rp * rp * rp - rm * rm * rm);
    const float det = cell[0] * (cell[4] * cell[8] - cell[5] * cell[7])
                    - cell[1] * (cell[3] * cell[8] - cell[5] * cell[6])
                    + cell[2] * (cell[3] * cell[7] - cell[4] * cell[6]);
    const float gr = h / v * det / (float)(N_ATOMS - 1) / (float)N_ATOMS * 2.0f;
    out[b] = r;
    out[nbins + b] = gr;
}

extern "C" void kernel_launch(void* const* d_in, const int* in_sizes, int n_in,
                              void* d_out, int out_size, void* d_ws, size_t ws_size,
                              hipStream_t stream) {
    const float* traj   = (const float*)d_in[0];
    const float* cell   = (const float*)d_in[1];
    const float* r_list = (const float*)d_in[2];
    const int nbins = in_sizes[2];             // 118
    float* ws = (float*)d_ws;                  // 128 f32 histogram accumulators

    rdf_zero_kernel<<<1, NBINS_PAD, 0, stream>>>(ws);
    rdf_hist_kernel<<<HIST_BLOCKS, BLOCK, 0, stream>>>(traj, cell, r_list, nbins, ws);
    rdf_finalize_kernel<<<(nbins + 127) / 128, 128, 0, stream>>>(r_list, cell, ws,
                                                                 (float*)d_out, nbins);
}